// WeightTransformBilateralFilter_38998303048132
// MI455X (gfx1250) — compile-verified
//
#include <hip/hip_runtime.h>
#include <stdint.h>

// Problem constants from setup_inputs(): x=(4,8,720,1280) f32, params=(10,),
// kernel_size=5, dilation=1, dynamic_size=3. Output=(4,3,720,1280) f32.
#define B_   4
#define C_   8
#define H_   720
#define W_   1280
#define OUTC 3

#define TILE_W 32
#define TILE_H 8
#define HALO   2
#define LW (TILE_W + 2 * HALO)        // 36
#define LH (TILE_H + 2 * HALO)        // 12
#define PLANE (LH * LW)               // 432 pixel slots
#define CHB   (H_ * W_ * 4)           // channel stride in bytes (3,686,400)

static __device__ __forceinline__ int clampi(int v, int lo, int hi) {
    return v < lo ? lo : (v > hi ? hi : v);   // -> v_med3_i32
}

// 25-tap loop; GUARD=false for tiles fully interior to the image.
template <bool GUARD>
static __device__ __forceinline__ void run_taps(
    const float4* __restrict__ smem4, const float* __restrict__ qmem,
    int lx, int ly, int gx, int gy, float qc,
    const float (&m)[C_], float sx, float sy,
    float& n0, float& n1, float& n2, float& den)
{
    #pragma unroll
    for (int i = 0; i < 5; ++i) {
        const int dy = i - 2;
        const bool yin = !GUARD || ((gy + dy) >= 0 && (gy + dy) < H_);
        const int rowb = (ly + dy) * LW + lx;          // one add per row
        const float4* arow = smem4 + rowb;
        const float4* brow = smem4 + (PLANE + rowb);
        const float*  qrow = qmem + rowb;
        #pragma unroll
        for (int j = 0; j < 5; ++j) {
            const int dx = j - 2;
            if (!GUARD || (yin && (gx + dx) >= 0 && (gx + dx) < W_)) {
                const float4 A  = arow[dx];            // ds_load_b128 (imm offs)
                const float4 Bv = brow[dx];            // ds_load_b128 (imm offs)
                const float  qt = qrow[dx];            // merges into 2addr loads

                const float spat = sx * (float)(dx * dx) + sy * (float)(dy * dy);
                float acc = qt + (qc + spat);
                acc = fmaf(m[0], A.x,  acc);
                acc = fmaf(m[1], A.y,  acc);
                acc = fmaf(m[2], A.z,  acc);
                acc = fmaf(m[3], A.w,  acc);
                acc = fmaf(m[4], Bv.x, acc);
                acc = fmaf(m[5], Bv.y, acc);
                acc = fmaf(m[6], Bv.z, acc);
                acc = fmaf(m[7], Bv.w, acc);

                const float w = __builtin_amdgcn_exp2f(acc);  // v_exp_f32
                n0  = fmaf(w, A.x, n0);
                n1  = fmaf(w, A.y, n1);
                n2  = fmaf(w, A.z, n2);
                den += w;
            }
        }
    }
}

__global__ __launch_bounds__(256)
void bilateral5x5_kernel(const float* __restrict__ x,
                         const float* __restrict__ params,
                         float* __restrict__ out)
{
    // smem4[g*PLANE + slot] holds channels 4g..4g+3 of pixel slot (row*LW+col).
    // qmem[slot] caches q(s) = sum_c tp_c * s_c^2 (pre-scaled by log2 e).
    __shared__ float4 smem4[2 * PLANE];   // 13824 B
    __shared__ float  qmem[PLANE];        //  1728 B

    const int tid = threadIdx.x;
    int bid = blockIdx.x;
    const int tx0 = (bid % 40) * TILE_W;
    bid /= 40;
    const int ty0 = (bid % 90) * TILE_H;
    const int b   = bid / 90;

    // ---------------- stage tile(+halo) into LDS: async GVS copies ----------
    // Plane-slot decode done exactly twice; per-channel copies are 2 adds each.
    const uint32_t lds_base = (uint32_t)(uintptr_t)(&smem4[0]);
    const uint64_t sbase    = (uint64_t)(uintptr_t)(x + (size_t)b * (C_ * H_ * W_));

    const uint32_t rem0 = (uint32_t)tid;          // plane slots 0..255
    const uint32_t r0   = rem0 / (uint32_t)LW;
    const uint32_t col0 = rem0 - r0 * (uint32_t)LW;
    const uint32_t rem1 = (uint32_t)tid + 256u;   // plane slots 256..431
    const uint32_t r1   = rem1 / (uint32_t)LW;
    const uint32_t col1 = rem1 - r1 * (uint32_t)LW;

    const int gy0 = clampi(ty0 - HALO + (int)r0, 0, H_ - 1);  // clamped halo;
    const int gx0 = clampi(tx0 - HALO + (int)col0, 0, W_ - 1);// out-of-image
    const int gy1 = clampi(ty0 - HALO + (int)r1, 0, H_ - 1);  // taps skipped
    const int gx1 = clampi(tx0 - HALO + (int)col1, 0, W_ - 1);// in compute
    const uint32_t rb0 = ((uint32_t)gy0 * W_ + (uint32_t)gx0) * 4u;
    const uint32_t rb1 = ((uint32_t)gy1 * W_ + (uint32_t)gx1) * 4u;
    const uint32_t ld0 = lds_base + rem0 * 16u;
    const uint32_t ld1 = lds_base + rem1 * 16u;

    #pragma unroll
    for (int c = 0; c < C_; ++c) {
        const uint32_t goff = (uint32_t)((c >> 2) * PLANE * 16 + (c & 3) * 4);
        uint32_t la = ld0 + goff;
        uint32_t vo = rb0 + (uint32_t)(c * CHB);
        asm volatile("global_load_async_to_lds_b32 %0, %1, %2"
                     :: "v"(la), "v"(vo), "s"(sbase) : "memory");
    }
    if (tid < (PLANE - 256)) {
        #pragma unroll
        for (int c = 0; c < C_; ++c) {
            const uint32_t goff = (uint32_t)((c >> 2) * PLANE * 16 + (c & 3) * 4);
            uint32_t la = ld1 + goff;
            uint32_t vo = rb1 + (uint32_t)(c * CHB);
            asm volatile("global_load_async_to_lds_b32 %0, %1, %2"
                         :: "v"(la), "v"(vo), "s"(sbase) : "memory");
        }
    }
    asm volatile("s_wait_asynccnt 0" ::: "memory");
    __syncthreads();

    // ---------------- uniform parameter transform (pre-scaled by log2 e) -----
    const float LOG2E = 1.4426950408889634f;
    float tp[C_];
    #pragma unroll
    for (int c = 0; c < C_; ++c) { float p = params[c]; tp[c] = -p * p * LOG2E; }
    float psx = params[C_];     const float sx = -psx * psx * LOG2E;
    float psy = params[C_ + 1]; const float sy = -psy * psy * LOG2E;

    // ---------------- per-pixel q(s) precompute into LDS ---------------------
    #pragma unroll
    for (int k = 0; k < 2; ++k) {
        int p = tid + k * 256;
        if (p < PLANE) {
            const float4 A  = smem4[p];
            const float4 Bv = smem4[PLANE + p];
            float q;
            q = (tp[0] * A.x)  * A.x;
            q = fmaf(tp[1] * A.y,  A.y,  q);
            q = fmaf(tp[2] * A.z,  A.z,  q);
            q = fmaf(tp[3] * A.w,  A.w,  q);
            q = fmaf(tp[4] * Bv.x, Bv.x, q);
            q = fmaf(tp[5] * Bv.y, Bv.y, q);
            q = fmaf(tp[6] * Bv.z, Bv.z, q);
            q = fmaf(tp[7] * Bv.w, Bv.w, q);
            qmem[p] = q;
        }
    }
    __syncthreads();

    // ---------------- 25-tap stencil: logw = q_tap + qc + dot(m, s) + spat ---
    const int txl = tid & (TILE_W - 1);
    const int tyl = tid >> 5;
    const int lx  = txl + HALO;
    const int ly  = tyl + HALO;
    const int gx  = tx0 + txl;
    const int gy  = ty0 + tyl;

    const int cidx  = ly * LW + lx;
    const float4 xA = smem4[cidx];
    const float4 xB = smem4[PLANE + cidx];
    const float  qc = qmem[cidx];

    float m[C_];  // m_c = -2 * tp_c * x_c
    m[0] = (-2.f * tp[0]) * xA.x;
    m[1] = (-2.f * tp[1]) * xA.y;
    m[2] = (-2.f * tp[2]) * xA.z;
    m[3] = (-2.f * tp[3]) * xA.w;
    m[4] = (-2.f * tp[4]) * xB.x;
    m[5] = (-2.f * tp[5]) * xB.y;
    m[6] = (-2.f * tp[6]) * xB.z;
    m[7] = (-2.f * tp[7]) * xB.w;

    float n0 = 0.f, n1 = 0.f, n2 = 0.f, den = 0.f;

    // Tile-interiority is block-uniform: 93% of blocks take the unguarded path.
    const bool interior = (tx0 >= HALO) && (tx0 + TILE_W + HALO <= W_) &&
                          (ty0 >= HALO) && (ty0 + TILE_H + HALO <= H_);
    if (interior) {
        run_taps<false>(smem4, qmem, lx, ly, gx, gy, qc, m, sx, sy, n0, n1, n2, den);
    } else {
        run_taps<true >(smem4, qmem, lx, ly, gx, gy, qc, m, sx, sy, n0, n1, n2, den);
    }

    // center tap contributes w ~= 1, so den >= ~1 -> fast rcp is safe
    const float inv = __builtin_amdgcn_rcpf(den);
    const size_t plane = (size_t)H_ * W_;
    size_t o = ((size_t)b * OUTC) * plane + (size_t)gy * W_ + gx;
    out[o]             = n0 * inv;
    out[o + plane]     = n1 * inv;
    out[o + 2 * plane] = n2 * inv;
}

extern "C" void kernel_launch(void* const* d_in, const int* in_sizes, int n_in,
                              void* d_out, int out_size, void* d_ws, size_t ws_size,
                              hipStream_t stream) {
    (void)in_sizes; (void)n_in; (void)d_ws; (void)ws_size; (void)out_size;
    const float* x      = (const float*)d_in[0];
    const float* params = (const float*)d_in[1];
    float* out          = (float*)d_out;

    dim3 grid(40 * 90 * B_);          // exact tiling: 1280/32 x 720/8 x 4
    bilateral5x5_kernel<<<grid, 256, 0, stream>>>(x, params, out);
}